// DetectionHead_2989297238029
// MI455X (gfx1250) — compile-verified
//
#include <hip/hip_runtime.h>
#include <math.h>

// ---------------------------------------------------------------------------
// DetectionHead for MI455X (gfx1250, wave32).
//   GEMM:  M=HW per batch, K=384, N=72(pad 80), fp32 via V_WMMA_F32_16X16X4_F32
//   TopK:  exact radix-select (4x 8-bit passes) + bitonic sort for sorted order
// Roofline: reading x (329MB) dominates -> ~14us @ 23.3TB/s; fp32 WMMA is free.
// ---------------------------------------------------------------------------

typedef __attribute__((ext_vector_type(2))) float v2f;
typedef __attribute__((ext_vector_type(8))) float v8f;

namespace {
constexpr int H = 248, W = 216, HW = H * W;        // 53568
constexpr int INCH = 384, BATCH = 4;
constexpr int NANCH = 6;                           // 3 classes x 2 dir bins
constexpr int NITEMS = HW * NANCH;                 // 321408 anchors per batch
constexpr int NPRE = 4096;                         // top-k
constexpr int CAP = 16384;                         // candidate buffer (pow2)
constexpr int NT = 5;                              // N-tiles of 16 -> 80 ch (72 + pad)
constexpr int KSTEPS = INCH / 4;                   // 96 K-steps of 4
constexpr int MTILES = HW / 16;                    // 3348 (divides exactly)
constexpr int WPB = 4;                             // waves per GEMM block

// workspace layout (bytes), all 256-aligned
constexpr size_t OFF_WPACK   = 0;                                        // 5*96*32 v2f = 122880B
constexpr size_t OFF_SKEYS   = 122880;                                   // B*N u32  = 5142528B
constexpr size_t OFF_PAYLOAD = OFF_SKEYS + (size_t)BATCH * NITEMS * 4;   // B*N*11 f32 = 56567808B
constexpr size_t OFF_HIST    = OFF_PAYLOAD + (size_t)BATCH * NITEMS * 44;// B*256 u32
constexpr size_t OFF_STATE   = OFF_HIST + BATCH * 256 * 4;               // prefix[B], kneed[B], cnt[B]
constexpr size_t OFF_CAND    = (OFF_STATE + 3 * BATCH * 4 + 255) & ~(size_t)255; // B*CAP u64
}

// --- clear per-call state / sentinel-fill candidate buffer ------------------
__global__ void init_ws(unsigned* hist, unsigned* prefix, unsigned* kneed,
                        unsigned* cnt, unsigned long long* cand) {
  int t = blockIdx.x * blockDim.x + threadIdx.x;
  for (int i = t; i < BATCH * CAP; i += blockDim.x * gridDim.x) cand[i] = 0ull; // sentinel
  if (t < BATCH * 256) hist[t] = 0u;
  if (t < BATCH) { prefix[t] = 0u; kneed[t] = (unsigned)NPRE; cnt[t] = 0u; }
}

// --- repack weights into the per-lane WMMA B-operand layout -----------------
// For f32 16x16x4 we assume B mirrors the documented A layout:
//   VGPR0: lanes 0-15 = (K=0, N=lane), lanes 16-31 = (K=2, N=lane-16)
//   VGPR1: lanes 0-15 = (K=1, N=lane), lanes 16-31 = (K=3, N=lane-16)
// wpack[(n*KSTEPS + k)*32 + lane] = {W[o][c], W[o][c+1]},
//   o = n*16 + (lane&15), c = 4k + 2*(lane>>4). Channels: 0-17 cls, 18-59 reg,
//   60-71 dir, 72-79 zero pad.
__global__ void pack_w(const float* __restrict__ wcls, const float* __restrict__ wreg,
                       const float* __restrict__ wdir, v2f* __restrict__ wpack) {
  int t = blockIdx.x * blockDim.x + threadIdx.x;
  if (t >= NT * KSTEPS * 32) return;
  int lane = t & 31;
  int k = (t >> 5) % KSTEPS;
  int n = t / (KSTEPS * 32);
  int o = n * 16 + (lane & 15);
  int c = 4 * k + 2 * (lane >> 4);
  v2f v; v.x = 0.0f; v.y = 0.0f;
  if (o < 18)      { v.x = wcls[o * INCH + c];        v.y = wcls[o * INCH + c + 1]; }
  else if (o < 60) { v.x = wreg[(o - 18) * INCH + c]; v.y = wreg[(o - 18) * INCH + c + 1]; }
  else if (o < 72) { v.x = wdir[(o - 60) * INCH + c]; v.y = wdir[(o - 60) * INCH + c + 1]; }
  wpack[t] = v;
}

// --- fused head GEMM + epilogue ---------------------------------------------
__global__ __launch_bounds__(128) void gemm_head(
    const float* __restrict__ x, const v2f* __restrict__ wpack,
    const float* __restrict__ bcls, const float* __restrict__ breg,
    const float* __restrict__ bdir,
    unsigned* __restrict__ skeys, float* __restrict__ payload) {
  __shared__ float ldsT[WPB][16][84];   // 84 stride: conflict-free transpose
  const int lane = threadIdx.x & 31;
  const int wv   = threadIdx.x >> 5;
  const int b    = blockIdx.y;
  const int mt   = blockIdx.x * WPB + wv;   // MTILES % WPB == 0, no tail
  const int p0   = mt * 16;
  const int m    = lane & 15;               // A: M index
  const int ks   = lane >> 4;               // A: K half-select (K= 2*ks / 2*ks+1)

  // A-operand element pointers (fp32, NCHW: channel stride = HW)
  const float* x0 = x + ((size_t)b * INCH + 2 * ks) * HW + p0 + m;
  const float* x1 = x0 + HW;
  const v2f*   wp = wpack + lane;

  v8f acc[NT];
  #pragma unroll
  for (int n = 0; n < NT; ++n)
    #pragma unroll
    for (int r = 0; r < 8; ++r) acc[n][r] = 0.0f;

  for (int k = 0; k < KSTEPS; ++k) {
    size_t off = (size_t)(4 * k) * HW;
    v2f a; a.x = x0[off]; a.y = x1[off];        // 2 coalesced B32 loads
    #pragma unroll
    for (int n = 0; n < NT; ++n) {
      v2f bb = wp[(n * KSTEPS + k) * 32];       // 1 coalesced b64 load (L2-hot)
      acc[n] = __builtin_amdgcn_wmma_f32_16x16x4_f32(
          /*neg_a=*/false, a, /*neg_b=*/false, bb,
          /*c_mod=*/(short)0, acc[n], /*reuse_a=*/false, /*reuse_b=*/false);
    }
  }

  // D layout: VGPR r, lane l -> M=(l>>4)*8+r, N=l&15. Transpose through LDS.
  #pragma unroll
  for (int n = 0; n < NT; ++n)
    #pragma unroll
    for (int r = 0; r < 8; ++r)
      ldsT[wv][ks * 8 + r][n * 16 + m] = acc[n][r];
  __syncthreads();

  // Epilogue: 16 positions x 6 anchors = 96 slots per wave
  for (int t = lane; t < 96; t += 32) {
    int pos = t / 6, a = t - 6 * (t / 6);
    const float* row = &ldsT[wv][pos][0];
    float z0 = row[a * 3 + 0] + bcls[a * 3 + 0];
    float z1 = row[a * 3 + 1] + bcls[a * 3 + 1];
    float z2 = row[a * 3 + 2] + bcls[a * 3 + 2];
    float rg[7];
    #pragma unroll
    for (int i = 0; i < 7; ++i) rg[i] = row[18 + a * 7 + i] + breg[a * 7 + i];
    float d0 = row[60 + a * 2 + 0] + bdir[a * 2 + 0];
    float d1 = row[60 + a * 2 + 1] + bdir[a * 2 + 1];

    float zmax = fmaxf(z0, fmaxf(z1, z2));      // sigmoid monotonic -> key on logit
    unsigned u = __float_as_uint(zmax);
    unsigned s = u ^ ((unsigned)((int)u >> 31) | 0x80000000u); // order-preserving

    size_t nidx = (size_t)(p0 + pos) * NANCH + a;
    size_t gi   = (size_t)b * NITEMS + nidx;
    skeys[gi] = s;
    float* pl = payload + gi * 11;
    pl[0] = 1.0f / (1.0f + expf(-z0));
    pl[1] = 1.0f / (1.0f + expf(-z1));
    pl[2] = 1.0f / (1.0f + expf(-z2));
    #pragma unroll
    for (int i = 0; i < 7; ++i) pl[3 + i] = rg[i];
    pl[10] = (d1 > d0) ? 1.0f : 0.0f;           // argmax, first-max ties -> 0
  }
}

// --- radix-select: histogram pass -------------------------------------------
__global__ void topk_hist(const unsigned* __restrict__ skeys,
                          const unsigned* __restrict__ prefix,
                          unsigned* __restrict__ hist, int pass) {
  __shared__ unsigned lh[256];
  int b = blockIdx.y;
  if (threadIdx.x < 256) lh[threadIdx.x] = 0u;
  __syncthreads();
  const unsigned* sk = skeys + (size_t)b * NITEMS;
  int shift = 24 - 8 * pass;
  unsigned pfx = prefix[b];
  unsigned maskhi = (pass == 0) ? 0u : (0xFFFFFFFFu << (shift + 8));
  for (int i = blockIdx.x * blockDim.x + threadIdx.x; i < NITEMS;
       i += blockDim.x * gridDim.x) {
    unsigned s = sk[i];
    if ((s & maskhi) == (pfx & maskhi)) atomicAdd(&lh[(s >> shift) & 0xFFu], 1u);
  }
  __syncthreads();
  if (threadIdx.x < 256 && lh[threadIdx.x])
    atomicAdd(&hist[b * 256 + threadIdx.x], lh[threadIdx.x]);
}

// --- radix-select: pick byte, update prefix/kneed, clear hist ---------------
__global__ void topk_scan(unsigned* hist, unsigned* prefix, unsigned* kneed, int pass) {
  int b = blockIdx.x;
  if (threadIdx.x == 0) {
    int shift = 24 - 8 * pass;
    unsigned need = kneed[b], cum = 0, pfx = prefix[b];
    for (int v = 255; v >= 0; --v) {
      unsigned c = hist[b * 256 + v];
      if (cum + c >= need) {
        prefix[b] = pfx | ((unsigned)v << shift);
        kneed[b]  = need - cum;
        break;
      }
      cum += c;
    }
  }
  __syncthreads();
  if (threadIdx.x < 256) hist[b * 256 + threadIdx.x] = 0u;
}

// --- compact everything >= exact threshold into composite keys --------------
__global__ void topk_compact(const unsigned* __restrict__ skeys,
                             const unsigned* __restrict__ prefix,
                             unsigned* __restrict__ cnt,
                             unsigned long long* __restrict__ cand) {
  int b = blockIdx.y;
  unsigned thr = prefix[b];
  const unsigned* sk = skeys + (size_t)b * NITEMS;
  for (int i = blockIdx.x * blockDim.x + threadIdx.x; i < NITEMS;
       i += blockDim.x * gridDim.x) {
    unsigned s = sk[i];
    if (s >= thr) {
      unsigned pos = atomicAdd(&cnt[b], 1u);
      if (pos < CAP)
        cand[(size_t)b * CAP + pos] =
            ((unsigned long long)s << 32) | (unsigned)(~(unsigned)i); // ~idx: ties -> smallest idx first
    }
  }
}

// --- per-batch bitonic sort, descending (sorted top-k like lax.top_k) -------
__global__ __launch_bounds__(1024) void topk_sort(unsigned long long* cand) {
  unsigned long long* a = cand + (size_t)blockIdx.x * CAP;
  for (int k = 2; k <= CAP; k <<= 1) {
    for (int j = k >> 1; j > 0; j >>= 1) {
      for (int i = threadIdx.x; i < CAP; i += 1024) {
        int p = i ^ j;
        if (p > i) {
          bool desc = ((i & k) == 0);
          unsigned long long ai = a[i], ap = a[p];
          if ((ai < ap) == desc) { a[i] = ap; a[p] = ai; }
        }
      }
      __syncthreads();
    }
  }
}

// --- gather + box decode -----------------------------------------------------
__global__ void finalize(const unsigned long long* __restrict__ cand,
                         const float* __restrict__ payload,
                         const float* __restrict__ anchors,
                         float* __restrict__ out) {
  int t = blockIdx.x * blockDim.x + threadIdx.x;
  if (t >= BATCH * NPRE) return;
  int b = t / NPRE, s = t - b * NPRE;
  unsigned long long c = cand[(size_t)b * CAP + s];
  unsigned idx = ~(unsigned)(c & 0xFFFFFFFFull);
  const float* pl = payload + ((size_t)b * NITEMS + idx) * 11;
  const float* a  = anchors + (size_t)idx * 7;
  float diag = sqrtf(a[3] * a[3] + a[4] * a[4]);
  float cx = pl[3] * diag + a[0];
  float cy = pl[4] * diag + a[1];
  float cz = pl[5] * a[5] + a[2] + 0.5f * a[5];
  float bw = a[3] * expf(pl[6]);
  float bl = a[4] * expf(pl[7]);
  float bh = a[5] * expf(pl[8]);
  cz -= 0.5f * bh;
  const float PI_ = 3.14159265358979323846f;     // PERIOD = 2*pi/2
  float dirf = pl[10];
  float ang = a[6] + pl[9];
  ang = ang - floorf(ang / PI_ + 1.0f) * PI_;
  ang = ang + (1.0f - dirf) * PI_;
  float* o = out + (size_t)t * 11;
  o[0] = cx; o[1] = cy; o[2] = cz; o[3] = bw; o[4] = bl; o[5] = bh; o[6] = ang;
  o[7] = pl[0]; o[8] = pl[1]; o[9] = pl[2]; o[10] = dirf;
}

extern "C" void kernel_launch(void* const* d_in, const int* in_sizes, int n_in,
                              void* d_out, int out_size, void* d_ws, size_t ws_size,
                              hipStream_t stream) {
  (void)in_sizes; (void)n_in; (void)out_size; (void)ws_size;
  const float* x    = (const float*)d_in[0];
  const float* anch = (const float*)d_in[1];
  const float* wcls = (const float*)d_in[2];
  const float* bcls = (const float*)d_in[3];
  const float* wreg = (const float*)d_in[4];
  const float* breg = (const float*)d_in[5];
  const float* wdir = (const float*)d_in[6];
  const float* bdir = (const float*)d_in[7];
  float* out = (float*)d_out;

  char* ws = (char*)d_ws;                    // requires ~62.4 MB workspace
  v2f*                wpack   = (v2f*)(ws + OFF_WPACK);
  unsigned*           skeys   = (unsigned*)(ws + OFF_SKEYS);
  float*              payload = (float*)(ws + OFF_PAYLOAD);
  unsigned*           hist    = (unsigned*)(ws + OFF_HIST);
  unsigned*           prefix  = (unsigned*)(ws + OFF_STATE);
  unsigned*           kneed   = prefix + BATCH;
  unsigned*           cnt     = prefix + 2 * BATCH;
  unsigned long long* cand    = (unsigned long long*)(ws + OFF_CAND);

  init_ws<<<256, 256, 0, stream>>>(hist, prefix, kneed, cnt, cand);
  pack_w<<<(NT * KSTEPS * 32 + 255) / 256, 256, 0, stream>>>(wcls, wreg, wdir, wpack);

  gemm_head<<<dim3(MTILES / WPB, BATCH), 128, 0, stream>>>(
      x, wpack, bcls, breg, bdir, skeys, payload);

  for (int pass = 0; pass < 4; ++pass) {
    topk_hist<<<dim3(256, BATCH), 256, 0, stream>>>(skeys, prefix, hist, pass);
    topk_scan<<<BATCH, 256, 0, stream>>>(hist, prefix, kneed, pass);
  }
  topk_compact<<<dim3(256, BATCH), 256, 0, stream>>>(skeys, prefix, cnt, cand);
  topk_sort<<<BATCH, 1024, 0, stream>>>(cand);
  finalize<<<(BATCH * NPRE + 255) / 256, 256, 0, stream>>>(cand, payload, anch, out);
}